// Spearman_39393440039365
// MI455X (gfx1250) — compile-verified
//
#include <hip/hip_runtime.h>
#include <math.h>

// ---------------------------------------------------------------------------
// Soft-rank Spearman correlation (torchsort-style, L2 regularization = 1.0)
// Pipeline per input vector (N = 262144, power of two):
//   1) bitonic sort descending (key=value f32, payload=original index)
//   2) PAVA isotonic regression on y_i = s_i - (n - i)   (sequential stack)
//   3) scatter rank[idx[i]] = (s_i - blockmean_i) * scale (binary search)
// Then Pearson correlation of the two rank vectors; the three dot products
// are accumulated with V_WMMA_F32_16X16X4_F32 (A = products, B = ones — the
// trace/row-sum trick makes the result independent of exact VGPR striping).
// ---------------------------------------------------------------------------

typedef __attribute__((ext_vector_type(2))) float v2f;
typedef __attribute__((ext_vector_type(8))) float v8f;

#define THREADS 256
#define MEAN_BLOCKS 1024
#define WMMA_BLOCKS 32

__global__ void k_init_sort(const float* __restrict__ vals, float* __restrict__ keys,
                            int* __restrict__ idx, int n) {
  int i = blockIdx.x * blockDim.x + threadIdx.x;
  if (i < n) { keys[i] = vals[i]; idx[i] = i; }
}

// One global bitonic pass for stride j (j >= 2048), stage k. n/2 threads.
__global__ void k_bitonic_global(float* __restrict__ keys, int* __restrict__ idx,
                                 int k, int j, int n) {
  int t = blockIdx.x * blockDim.x + threadIdx.x;
  int p = ((t & ~(j - 1)) << 1) | (t & (j - 1));
  int q = p | j;
  if (q < n) {
    bool desc = ((p & k) == 0);          // overall DESCENDING sort
    float a = keys[p], b = keys[q];
    if ((a < b) == desc) {
      keys[p] = b; keys[q] = a;
      int ia = idx[p]; idx[p] = idx[q]; idx[q] = ia;
    }
  }
}

// All passes with stride <= 1024 for stage k, fused in LDS over 2048 elements.
__global__ void k_bitonic_lds(float* __restrict__ keys, int* __restrict__ idx,
                              int k, int jstart) {
  __shared__ float sk[2048];
  __shared__ int   si[2048];
  int base = blockIdx.x * 2048;
  int t = threadIdx.x;                    // 1024 threads
  sk[t]        = keys[base + t];        si[t]        = idx[base + t];
  sk[t + 1024] = keys[base + t + 1024]; si[t + 1024] = idx[base + t + 1024];
  __syncthreads();
  for (int j = jstart; j >= 1; j >>= 1) {
    int p = ((t & ~(j - 1)) << 1) | (t & (j - 1));
    int q = p | j;
    bool desc = (((base + p) & k) == 0);
    float a = sk[p], b = sk[q];
    if ((a < b) == desc) {
      sk[p] = b; sk[q] = a;
      int ia = si[p]; si[p] = si[q]; si[q] = ia;
    }
    __syncthreads();
  }
  keys[base + t]        = sk[t];        idx[base + t]        = si[t];
  keys[base + t + 1024] = sk[t + 1024]; idx[base + t + 1024] = si[t + 1024];
}

// Sequential Pool-Adjacent-Violators on y_i = s_i - (n - i) (non-increasing fit).
// Matches the reference merge test: sums[t-2]*cnt[t-1] <= sums[t-1]*cnt[t-2].
__global__ void k_pava_build(const float* __restrict__ skeys, float* __restrict__ sums,
                             float* __restrict__ cnts, int* __restrict__ starts,
                             int* __restrict__ top_out, int n) {
  if (threadIdx.x != 0 || blockIdx.x != 0) return;
  int top = 0;
  for (int i = 0; i < n; ++i) {
    if ((i & 63) == 0) __builtin_prefetch(&skeys[i + 256], 0, 0);
    float y = skeys[i] - (float)(n - i);
    sums[top] = y; cnts[top] = 1.0f; starts[top] = i; ++top;
    while (top > 1 && sums[top - 2] * cnts[top - 1] <= sums[top - 1] * cnts[top - 2]) {
      sums[top - 2] += sums[top - 1];
      cnts[top - 2] += cnts[top - 1];
      --top;
    }
  }
  *top_out = top;
}

// rank[idx[i]] = (s_i - mean(block containing i)) * scale ; block via bsearch.
__global__ void k_scatter_ranks(const float* __restrict__ skeys, const int* __restrict__ idx,
                                const float* __restrict__ sums, const float* __restrict__ cnts,
                                const int* __restrict__ starts, const int* __restrict__ top_p,
                                float* __restrict__ rank, float scale, int n) {
  int i = blockIdx.x * blockDim.x + threadIdx.x;
  if (i >= n) return;
  int top = *top_p;
  int lo = 0, hi = top - 1;
  while (lo < hi) {
    int mid = (lo + hi + 1) >> 1;
    if (starts[mid] <= i) lo = mid; else hi = mid - 1;
  }
  float mean = sums[lo] / fmaxf(cnts[lo], 1.0f);
  rank[idx[i]] = (skeys[i] - mean) * scale;
}

// Deterministic two-stage mean reduction (fixed-order LDS trees, no atomics).
__global__ void k_partial_means(const float* __restrict__ rp, const float* __restrict__ rt,
                                float* __restrict__ partsP, float* __restrict__ partsT, int n) {
  __shared__ float sp[THREADS], st[THREADS];
  float p = 0.0f, t = 0.0f;
  for (int j = blockIdx.x * blockDim.x + threadIdx.x; j < n; j += gridDim.x * blockDim.x) {
    p += rp[j]; t += rt[j];
  }
  sp[threadIdx.x] = p; st[threadIdx.x] = t;
  __syncthreads();
  for (int s = THREADS / 2; s > 0; s >>= 1) {
    if (threadIdx.x < (unsigned)s) {
      sp[threadIdx.x] += sp[threadIdx.x + s];
      st[threadIdx.x] += st[threadIdx.x + s];
    }
    __syncthreads();
  }
  if (threadIdx.x == 0) { partsP[blockIdx.x] = sp[0]; partsT[blockIdx.x] = st[0]; }
}

__global__ void k_reduce_means(const float* __restrict__ partsP, const float* __restrict__ partsT,
                               float* __restrict__ means, int nparts, int n) {
  __shared__ float sp[THREADS], st[THREADS];
  float p = 0.0f, t = 0.0f;
  for (int j = threadIdx.x; j < nparts; j += blockDim.x) { p += partsP[j]; t += partsT[j]; }
  sp[threadIdx.x] = p; st[threadIdx.x] = t;
  __syncthreads();
  for (int s = THREADS / 2; s > 0; s >>= 1) {
    if (threadIdx.x < (unsigned)s) {
      sp[threadIdx.x] += sp[threadIdx.x + s];
      st[threadIdx.x] += st[threadIdx.x + s];
    }
    __syncthreads();
  }
  if (threadIdx.x == 0) { means[0] = sp[0] / (float)n; means[1] = st[0] / (float)n; }
}

// Centered dot products via f32 WMMA (16x16x4). A rows carry 4 consecutive
// elementwise products, B = all ones => every D entry is a row partial-sum;
// summing the whole 16x16 accumulator yields 16x the chunk sum, independent
// of the exact A/B lane striping. The main loop is unguarded (full 64-elem
// tiles only, float2/global_load_b64 loads, EXEC stays all-ones with no
// save/restore), so loads batch and pipeline ahead of the three WMMAs.
// Any non-multiple-of-64 tail is handled scalar by wave 0 after the WMMAs.
__global__ void k_wmma_dots(const float* __restrict__ rp, const float* __restrict__ rt,
                            const float* __restrict__ means,
                            float* __restrict__ partsPP, float* __restrict__ partsTT,
                            float* __restrict__ partsPT, int n) {
  __builtin_amdgcn_s_wait_tensorcnt((short)0);   // CDNA5 split-counter wait (no-op here)
  const float mp = means[0], mt = means[1];
  const int lane = threadIdx.x & 31;
  const int wavesPerBlock = blockDim.x >> 5;
  const int waveId = blockIdx.x * wavesPerBlock + (threadIdx.x >> 5);
  const int numWaves = gridDim.x * wavesPerBlock;
  const int q = lane & 15;
  const int koff2 = lane >> 4;                   // lanes 0-15 -> K{0,1}; 16-31 -> K{2,3}

  const float2* __restrict__ rp2 = (const float2*)rp;
  const float2* __restrict__ rt2 = (const float2*)rt;

  v8f accPP = {}; v8f accTT = {}; v8f accPT = {};
  v2f ones; ones.x = 1.0f; ones.y = 1.0f;

  const int nFull = n & ~63;                     // whole 64-element tiles
  const int stride2 = numWaves * 32;             // tile stride in float2 units
  for (int base2 = waveId * 32; base2 < (nFull >> 1); base2 += stride2) {
    int i2 = base2 + q * 2 + koff2;              // float2 index of this lane's pair
    float2 pv = rp2[i2];
    float2 tv = rt2[i2];
    float p0 = pv.x - mp, p1 = pv.y - mp;
    float t0 = tv.x - mt, t1 = tv.y - mt;
    v2f aPP; aPP.x = p0 * p0; aPP.y = p1 * p1;
    v2f aTT; aTT.x = t0 * t0; aTT.y = t1 * t1;
    v2f aPT; aPT.x = p0 * t0; aPT.y = p1 * t1;
    accPP = __builtin_amdgcn_wmma_f32_16x16x4_f32(false, aPP, false, ones, (short)0, accPP, false, false);
    accTT = __builtin_amdgcn_wmma_f32_16x16x4_f32(false, aTT, false, ones, (short)0, accTT, false, false);
    accPT = __builtin_amdgcn_wmma_f32_16x16x4_f32(false, aPT, false, ones, (short)0, accPT, false, false);
  }

  float sPP = 0.0f, sTT = 0.0f, sPT = 0.0f;
  for (int e = 0; e < 8; ++e) { sPP += accPP[e]; sTT += accTT[e]; sPT += accPT[e]; }
  sPP *= (1.0f / 16.0f);                         // 16 columns replicate each row sum
  sTT *= (1.0f / 16.0f);
  sPT *= (1.0f / 16.0f);

  // Scalar tail for n % 64 != 0 (no-op for N = 262144); WMMAs all retired.
  if (waveId == 0) {
    for (int i = nFull + lane; i < n; i += 32) {
      float p = rp[i] - mp, t = rt[i] - mt;
      sPP += p * p; sTT += t * t; sPT += p * t;
    }
  }

  __shared__ float sh0[THREADS], sh1[THREADS], sh2[THREADS];
  sh0[threadIdx.x] = sPP; sh1[threadIdx.x] = sTT; sh2[threadIdx.x] = sPT;
  __syncthreads();
  for (int s = THREADS / 2; s > 0; s >>= 1) {
    if (threadIdx.x < (unsigned)s) {
      sh0[threadIdx.x] += sh0[threadIdx.x + s];
      sh1[threadIdx.x] += sh1[threadIdx.x + s];
      sh2[threadIdx.x] += sh2[threadIdx.x + s];
    }
    __syncthreads();
  }
  if (threadIdx.x == 0) {
    partsPP[blockIdx.x] = sh0[0];
    partsTT[blockIdx.x] = sh1[0];
    partsPT[blockIdx.x] = sh2[0];
  }
}

__global__ void k_final(const float* __restrict__ partsPP, const float* __restrict__ partsTT,
                        const float* __restrict__ partsPT, int nparts, float* __restrict__ out) {
  __shared__ float sh0[64], sh1[64], sh2[64];
  int t = threadIdx.x;                     // 64 threads
  float a = 0.0f, b = 0.0f, c = 0.0f;
  for (int j = t; j < nparts; j += blockDim.x) { a += partsPP[j]; b += partsTT[j]; c += partsPT[j]; }
  sh0[t] = a; sh1[t] = b; sh2[t] = c;
  __syncthreads();
  for (int s = 32; s > 0; s >>= 1) {
    if (t < s) { sh0[t] += sh0[t + s]; sh1[t] += sh1[t + s]; sh2[t] += sh2[t + s]; }
    __syncthreads();
  }
  if (t == 0) {
    float denom = sqrtf(sh0[0]) * sqrtf(sh1[0]);
    out[0] = (denom > 0.0f) ? (sh2[0] / denom) : 0.0f;
  }
}

extern "C" void kernel_launch(void* const* d_in, const int* in_sizes, int n_in,
                              void* d_out, int out_size, void* d_ws, size_t ws_size,
                              hipStream_t stream) {
  const float* target = (const float*)d_in[0];
  const float* pred   = (const float*)d_in[1];
  const int n = in_sizes[0];               // 262144 (power of two, >= 2048)
  float* out = (float*)d_out;

  char* ws = (char*)d_ws;
  const size_t fN = (size_t)n * sizeof(float);
  float* keys    = (float*)(ws + 0 * fN);
  int*   idx     = (int*)  (ws + 1 * fN);
  float* rank_p  = (float*)(ws + 2 * fN);
  float* rank_t  = (float*)(ws + 3 * fN);
  float* psums   = (float*)(ws + 4 * fN);
  float* pcnts   = (float*)(ws + 5 * fN);
  int*   pstarts = (int*)  (ws + 6 * fN);
  float* extra   = (float*)(ws + 7 * fN);
  int*   top     = (int*)extra;            // [0]
  float* means   = extra + 8;              // [8..9]
  float* partsP  = extra + 16;
  float* partsT  = partsP + MEAN_BLOCKS;
  float* partsPP = partsT + MEAN_BLOCKS;
  float* partsTT = partsPP + WMMA_BLOCKS;
  float* partsPT = partsTT + WMMA_BLOCKS;
  // total ws usage: 7*N*4 + ~9 KB  (~7.35 MB for N = 262144)

  const int blocks = (n + THREADS - 1) / THREADS;
  const int halfBlocks = (n / 2 + THREADS - 1) / THREADS;

  auto run_softrank = [&](const float* vals, float* rank, float scale) {
    k_init_sort<<<blocks, THREADS, 0, stream>>>(vals, keys, idx, n);
    for (int kk = 2; kk <= n; kk <<= 1) {
      int j = kk >> 1;
      for (; j >= 2048; j >>= 1)
        k_bitonic_global<<<halfBlocks, THREADS, 0, stream>>>(keys, idx, kk, j, n);
      k_bitonic_lds<<<n / 2048, 1024, 0, stream>>>(keys, idx, kk, j); // j <= 1024
    }
    k_pava_build<<<1, 32, 0, stream>>>(keys, psums, pcnts, pstarts, top, n);
    k_scatter_ranks<<<blocks, THREADS, 0, stream>>>(keys, idx, psums, pcnts, pstarts,
                                                    top, rank, scale, n);
  };

  run_softrank(pred,   rank_p, 1.0f / (float)n);  // pred_r / n (scale-invariant, kept for fidelity)
  run_softrank(target, rank_t, 1.0f);

  k_partial_means<<<MEAN_BLOCKS, THREADS, 0, stream>>>(rank_p, rank_t, partsP, partsT, n);
  k_reduce_means<<<1, THREADS, 0, stream>>>(partsP, partsT, means, MEAN_BLOCKS, n);
  k_wmma_dots<<<WMMA_BLOCKS, THREADS, 0, stream>>>(rank_p, rank_t, means,
                                                   partsPP, partsTT, partsPT, n);
  k_final<<<1, 64, 0, stream>>>(partsPP, partsTT, partsPT, WMMA_BLOCKS, out);
}